// Mamba3Triton_38628935860575
// MI455X (gfx1250) — compile-verified
//
#include <hip/hip_runtime.h>
#include <hip/hip_bf16.h>
#include <math.h>

// ---------------------------------------------------------------------------
// Problem constants (from reference)
// ---------------------------------------------------------------------------
#define BATCH 2
#define SEQ   256
#define DM    768
#define HD    64
#define NST   64
#define DI    1536           // EXPAND*DM
#define NH    24             // DI/HD
#define RR    2              // rank R
#define NA    16             // SPLIT/2
#define SPLIT 32
#define DINP  3416
#define MROWS (BATCH*SEQ)    // 512

// proj row offsets
#define OFF_Z    0
#define OFF_X    1536
#define OFF_B    3072
#define OFF_C    3200
#define OFF_DT   3328
#define OFF_A    3352
#define OFF_TRAP 3376
#define OFF_ANG  3400

typedef __attribute__((ext_vector_type(2))) float v2f;
typedef __attribute__((ext_vector_type(8))) float v8f;

__device__ __forceinline__ float softplus_f(float x) {
    return (x > 20.0f) ? x : log1pf(expf(x));
}
__device__ __forceinline__ float sigmoid_f(float x) {
    return 1.0f / (1.0f + expf(-x));
}

// ---------------------------------------------------------------------------
// fp32 WMMA GEMM:  C[M,N] = A[M,K] * W[N,K]^T    (one wave per 16x16 tile)
// V_WMMA_F32_16X16X4_F32: A 16x4 (2 VGPR), B 4x16 (2 VGPR), C/D 16x16 (8 VGPR)
// A layout: lanes 0-15 -> M=lane, K={0,1}; lanes 16-31 -> M=lane-16, K={2,3}
// B layout: lanes 0-15 -> N=lane, K={0,1}; lanes 16-31 -> N=lane-16, K={2,3}
// C layout: VGPR i: lanes 0-15 -> M=i, N=lane; lanes 16-31 -> M=8+i, N=lane-16
// ---------------------------------------------------------------------------
__global__ void gemm_f32_wmma(const float* __restrict__ A,
                              const float* __restrict__ W,
                              float* __restrict__ Cout,
                              int M, int N, int K) {
    const int nt   = blockIdx.x;
    const int mt   = blockIdx.y;
    const int lane = threadIdx.x;      // 0..31, full wave (EXEC all ones)
    const int half = lane >> 4;        // 0 or 1
    const int lrow = lane & 15;

    const int m = mt * 16 + lrow;                 // A row for this lane
    int n = nt * 16 + lrow;                       // W row (output col)
    const int nc = (n < N) ? n : (N - 1);         // clamp tail reads

    const float* __restrict__ arow = A + (size_t)m  * K + 2 * half;
    const float* __restrict__ wrow = W + (size_t)nc * K + 2 * half;

    v8f acc = {};
    #pragma unroll 8
    for (int k = 0; k < K; k += 4) {
        if ((k & 127) == 0) {
            __builtin_prefetch(arow + k + 512, 0, 0);
            __builtin_prefetch(wrow + k + 512, 0, 0);
        }
        v2f av = *reinterpret_cast<const v2f*>(arow + k);
        v2f bv = *reinterpret_cast<const v2f*>(wrow + k);
        acc = __builtin_amdgcn_wmma_f32_16x16x4_f32(
            /*neg_a=*/false, av, /*neg_b=*/false, bv,
            /*c_mod=*/(short)0, acc, /*reuse_a=*/false, /*reuse_b=*/false);
    }

    const int ccol = nt * 16 + lrow;
    if (ccol < N) {
        float* crow = Cout + (size_t)(mt * 16 + half * 8) * N + ccol;
        #pragma unroll
        for (int i = 0; i < 8; ++i)
            crow[(size_t)i * N] = acc[i];
    }
}

// ---------------------------------------------------------------------------
// Prep: per (b,l) row -> gate scalars DT/at/lam + RMS norms of Bm, Cm
// block 256 threads, grid = 512 rows
// ---------------------------------------------------------------------------
__global__ void prep_kernel(const float* __restrict__ proj,
                            const float* __restrict__ dt_bias,
                            const float* __restrict__ Bnw,
                            const float* __restrict__ Cnw,
                            float* __restrict__ DTa,
                            float* __restrict__ ata,
                            float* __restrict__ lama,
                            float* __restrict__ Bn,
                            float* __restrict__ Cn) {
    const int row = blockIdx.x;
    const int tid = threadIdx.x;
    const float* prow = proj + (size_t)row * DINP;
    __shared__ float red[256];

    if (tid < NH) {
        float DTv  = softplus_f(prow[OFF_DT + tid] + dt_bias[tid]);
        float Av   = -fmaxf(softplus_f(prow[OFF_A + tid]), 1e-4f);
        DTa[row * NH + tid]  = DTv;
        ata[row * NH + tid]  = expf(Av * DTv);
        lama[row * NH + tid] = sigmoid_f(prow[OFF_TRAP + tid]);
    }

    // 4 segments of 64: Bm r0, Bm r1, Cm r0, Cm r1
    const int seg = tid >> 6;
    const int idx = tid & 63;
    float val = (seg < 2) ? prow[OFF_B + seg * 64 + idx]
                          : prow[OFF_C + (seg - 2) * 64 + idx];
    red[tid] = val * val;
    __syncthreads();
    for (int off = 32; off > 0; off >>= 1) {
        if (idx < off) red[tid] += red[tid + off];
        __syncthreads();
    }
    float scale = rsqrtf(red[seg * 64] * (1.0f / 64.0f) + 1e-5f);
    float w = (seg < 2) ? Bnw[idx] : Cnw[idx];
    float o = val * scale * w;
    if (seg < 2) Bn[((size_t)row * RR + seg) * NST + idx] = o;
    else         Cn[((size_t)row * RR + (seg - 2)) * NST + idx] = o;
}

// ---------------------------------------------------------------------------
// Theta cumsum: grid = (b*NH), 16 threads (one per NA index), serial over L
// ---------------------------------------------------------------------------
__global__ void theta_kernel(const float* __restrict__ proj,
                             const float* __restrict__ DTa,
                             float* __restrict__ cth,
                             float* __restrict__ sth) {
    const int b = blockIdx.x / NH;
    const int h = blockIdx.x % NH;
    const int j = threadIdx.x;          // 0..15
    float th = 0.0f;
    for (int l = 0; l < SEQ; ++l) {
        int row = b * SEQ + l;
        th += DTa[row * NH + h] * proj[(size_t)row * DINP + OFF_ANG + j];
        cth[((size_t)row * NH + h) * NA + j] = cosf(th);
        sth[((size_t)row * NH + h) * NA + j] = sinf(th);
    }
}

// ---------------------------------------------------------------------------
// RoPE + bias: Bh/Ch[row, r, h, n] = rope(Bn[row,r,:] * bias[h,r,:])
// grid = 512 rows, block 256; 6144 elements per row (B and C)
// ---------------------------------------------------------------------------
__global__ void rope_kernel(const float* __restrict__ Bn,
                            const float* __restrict__ Cn,
                            const float* __restrict__ Bb,
                            const float* __restrict__ Cb,
                            const float* __restrict__ cth,
                            const float* __restrict__ sth,
                            float* __restrict__ Bh,
                            float* __restrict__ Ch) {
    const int row = blockIdx.x;
    for (int it = 0; it < 24; ++it) {              // 2*RR*NH*NST / 256
        int idx = it * 256 + threadIdx.x;
        int mat = idx / (RR * NH * NST);
        int rem = idx % (RR * NH * NST);
        int r   = rem / (NH * NST);
        int rem2 = rem % (NH * NST);
        int h = rem2 / NST;
        int n = rem2 % NST;

        const float* base = (mat == 0) ? Bn : Cn;
        const float* bias = (mat == 0) ? Bb : Cb;
        const size_t bofs  = ((size_t)row * RR + r) * NST;
        const size_t biofs = ((size_t)h * RR + r) * NST;   // bias is (NH,R,NST)

        float out;
        if (n < SPLIT) {
            int a = n >> 1;
            float c = cth[((size_t)row * NH + h) * NA + a];
            float s = sth[((size_t)row * NH + h) * NA + a];
            float v0 = base[bofs + 2 * a]     * bias[biofs + 2 * a];
            float v1 = base[bofs + 2 * a + 1] * bias[biofs + 2 * a + 1];
            out = (n & 1) ? (v0 * s + v1 * c) : (v0 * c - v1 * s);
        } else {
            out = base[bofs + n] * bias[biofs + n];
        }
        float* dst = (mat == 0) ? Bh : Ch;
        dst[(((size_t)row * RR + r) * NH + h) * NST + n] = out;
    }
}

// ---------------------------------------------------------------------------
// Sequential SSM scan: one block per (b,h); 128 threads = (r, p)
// Each thread holds S[p][0..63] and previous B row in VGPRs.
// ---------------------------------------------------------------------------
__global__ void __launch_bounds__(128)
scan_kernel(const float* __restrict__ proj,
            const float* __restrict__ Bh,
            const float* __restrict__ Ch,
            const float* __restrict__ DTa,
            const float* __restrict__ ata,
            const float* __restrict__ lama,
            const float* __restrict__ mimo_x,   // (NH, R, HD)
            const float* __restrict__ mimo_o,
            const float* __restrict__ Dv,
            float* __restrict__ Y) {
    const int b = blockIdx.x / NH;
    const int h = blockIdx.x % NH;
    const int tid = threadIdx.x;
    const int r = tid >> 6;
    const int p = tid & 63;

    const float mv = mimo_x[((size_t)h * RR + r) * HD + p];
    const float mo = mimo_o[((size_t)h * RR + r) * HD + p];
    const float Dh = Dv[h];

    float S[NST];
    float Bp[NST];
    #pragma unroll
    for (int i = 0; i < NST; ++i) { S[i] = 0.0f; Bp[i] = 0.0f; }
    float vprev = 0.0f;

    __shared__ float sB[2 * NST];
    __shared__ float sC[2 * NST];
    __shared__ float sScal[3];
    __shared__ float sY[128];

    for (int t = 0; t < SEQ; ++t) {
        const int row = b * SEQ + t;
        __syncthreads();   // protect sB/sC/sScal/sY from previous iteration
        const size_t bc = (((size_t)row * RR + r) * NH + h) * NST + p;
        sB[tid] = Bh[bc];
        sC[tid] = Ch[bc];
        if (tid == 0) {
            sScal[0] = ata[row * NH + h];
            sScal[1] = DTa[row * NH + h];
            sScal[2] = lama[row * NH + h];
        }
        const float vt = proj[(size_t)row * DINP + OFF_X + h * HD + p];
        const float zv = proj[(size_t)row * DINP + OFF_Z + h * HD + p];
        __syncthreads();

        const float at  = sScal[0];
        const float DTv = sScal[1];
        const float lam = sScal[2];
        const float coefc = DTv * lam * (vt * mv);
        const float coefp = DTv * (1.0f - lam) * at * (vprev * mv);

        const float* bptr = &sB[r * NST];
        const float* cptr = &sC[r * NST];
        float acc = 0.0f;
        #pragma unroll
        for (int n = 0; n < NST; ++n) {
            float bn = bptr[n];
            float sn = at * S[n] + coefc * bn + coefp * Bp[n];
            acc += cptr[n] * sn;
            S[n]  = sn;
            Bp[n] = bn;
        }
        sY[tid] = acc * mo;
        __syncthreads();
        if (r == 0) {
            float y = sY[p] + sY[64 + p] + Dh * vt;
            Y[(size_t)row * DI + h * HD + p] = y * (zv * sigmoid_f(zv));
        }
        vprev = vt;
    }
}

// ---------------------------------------------------------------------------
// Host-side launch
// ---------------------------------------------------------------------------
extern "C" void kernel_launch(void* const* d_in, const int* in_sizes, int n_in,
                              void* d_out, int out_size, void* d_ws, size_t ws_size,
                              hipStream_t stream) {
    const float* u        = (const float*)d_in[0];   // (2,256,768)
    const float* in_w     = (const float*)d_in[1];   // (3416,768)
    const float* dt_bias  = (const float*)d_in[2];   // (24,)
    const float* B_bias   = (const float*)d_in[3];   // (24,2,64)
    const float* C_bias   = (const float*)d_in[4];   // (24,2,64)
    const float* B_norm   = (const float*)d_in[5];   // (64,)
    const float* C_norm   = (const float*)d_in[6];   // (64,)
    const float* mimo_x   = (const float*)d_in[7];   // (24,2,64)
    const float* mimo_o   = (const float*)d_in[8];   // (24,2,64)
    const float* Dv       = (const float*)d_in[9];   // (24,)
    const float* out_w    = (const float*)d_in[10];  // (768,1536)
    float* out = (float*)d_out;                      // (2,256,768)

    // Workspace carve-up (floats)
    float* ws   = (float*)d_ws;
    float* proj = ws;                                  // 512*3416
    float* DTa  = proj + (size_t)MROWS * DINP;         // 512*24
    float* ata  = DTa  + (size_t)MROWS * NH;
    float* lama = ata  + (size_t)MROWS * NH;
    float* Bn   = lama + (size_t)MROWS * NH;           // 512*2*64
    float* Cn   = Bn   + (size_t)MROWS * RR * NST;
    float* cth  = Cn   + (size_t)MROWS * RR * NST;     // 512*24*16
    float* sth  = cth  + (size_t)MROWS * NH * NA;
    float* Bh   = sth  + (size_t)MROWS * NH * NA;      // 512*2*24*64
    float* Ch   = Bh   + (size_t)MROWS * RR * NH * NST;
    float* Yb   = Ch   + (size_t)MROWS * RR * NH * NST; // 512*1536

    // 1) in_proj GEMM: proj[512,3416] = u[512,768] @ in_w^T
    {
        dim3 grid((DINP + 15) / 16, MROWS / 16);
        gemm_f32_wmma<<<grid, dim3(32), 0, stream>>>(u, in_w, proj,
                                                     MROWS, DINP, DM);
    }
    // 2) gates + RMS norms
    prep_kernel<<<MROWS, 256, 0, stream>>>(proj, dt_bias, B_norm, C_norm,
                                           DTa, ata, lama, Bn, Cn);
    // 3) RoPE phase cumsum
    theta_kernel<<<BATCH * NH, NA, 0, stream>>>(proj, DTa, cth, sth);
    // 4) bias + RoPE -> Bh, Ch
    rope_kernel<<<MROWS, 256, 0, stream>>>(Bn, Cn, B_bias, C_bias,
                                           cth, sth, Bh, Ch);
    // 5) sequential scan -> Y
    scan_kernel<<<BATCH * NH, 128, 0, stream>>>(proj, Bh, Ch, DTa, ata, lama,
                                                mimo_x, mimo_o, Dv, Yb);
    // 6) out_proj GEMM: out[512,768] = Y[512,1536] @ out_w^T
    {
        dim3 grid(DM / 16, MROWS / 16);
        gemm_f32_wmma<<<grid, dim3(32), 0, stream>>>(Yb, out_w, out,
                                                     MROWS, DM, DI);
    }
}